// RecurrentGNN_80075370266744
// MI455X (gfx1250) — compile-verified
//
#include <hip/hip_runtime.h>
#include <math.h>

typedef __attribute__((ext_vector_type(2))) float v2f;
typedef __attribute__((ext_vector_type(8))) float v8f;
typedef __attribute__((ext_vector_type(4))) unsigned int u32x4;
typedef __attribute__((ext_vector_type(4))) int i32x4;
typedef __attribute__((ext_vector_type(8))) int i32x8;

#define HIDC  128
#define OUTC  64
#define A_STRIDE 132     // 16x128 A tile padded: bank = (4*r + k) % 64, conflict-free
#define LDS_STRIDE 516   // gates tile: rows m and m+8 land in different banks

#if defined(__has_builtin)
#if __has_builtin(__builtin_amdgcn_tensor_load_to_lds) && \
    __has_builtin(__builtin_amdgcn_s_wait_tensorcnt)
#define USE_TDM 1
#endif
#endif

static __device__ __forceinline__ float sigmoidf_(float x) {
  return 1.0f / (1.0f + __expf(-x));
}

// ---------------------------------------------------------------------------
// Fused LSTM layer, seq_len=1, zero state:
//   gates = inp @ wih^T + bih + bhh ;  c = sig(i)*tanh(g) ; out = sig(o)*tanh(c)
// Block = 512 threads (16 waves). Biases pulled into LDS by the Tensor Data
// Mover (tensor_load_to_lds + s_wait_tensorcnt); A tile staged in LDS;
// gates staged in LDS; then activations.
// ---------------------------------------------------------------------------
__global__ void lstm_layer_kernel(const float* __restrict__ inp,
                                  const float* __restrict__ wih,   // [512][128]
                                  const float* __restrict__ bih,   // [512]
                                  const float* __restrict__ bhh,   // [512]
                                  float* __restrict__ out) {
  __shared__ float bias_lds[1024];          // declared first -> LDS offset 0
  __shared__ float As[16 * A_STRIDE];
  __shared__ float gl[16 * LDS_STRIDE];

  const int lane = threadIdx.x & 31;
  const int wave = threadIdx.x >> 5;
  const int row0 = blockIdx.x << 4;
  const int r    = lane & 15;
  const int h2   = lane >> 4;

#ifdef USE_TDM
  if (wave == 0) {
    // D# group1: data_size=4B(2), tensor_dim0=512, tensor_dim1=1,
    // tile_dim0=512, tile_dim1=1, tensor_dim0_stride=512  (1-D tile)
    const unsigned long long q0 = (2ull << 16) | (512ull << 48);
    const unsigned long long q1 = (1ull << 16) | (512ull << 48);
    const unsigned long long q2 = 1ull | (512ull << 32);
    i32x8 g1;
    g1[0] = (int)(unsigned)q0; g1[1] = (int)(q0 >> 32);
    g1[2] = (int)(unsigned)q1; g1[3] = (int)(q1 >> 32);
    g1[4] = (int)(unsigned)q2; g1[5] = (int)(q2 >> 32);
    g1[6] = 0;                 g1[7] = 0;
    const i32x4 gz4 = {0, 0, 0, 0};
    const i32x8 gz8 = {0, 0, 0, 0, 0, 0, 0, 0};
    // D# group0: count=1, lds_addr, global_addr[56:0], type=2 (bits 127:126)
    const unsigned long long ga = (unsigned long long)(uintptr_t)bih;
    u32x4 g0;
    g0.x = 1u;
    g0.y = 0u;                                   // LDS byte offset of bias_lds
    g0.z = (unsigned)(ga & 0xFFFFFFFFu);
    g0.w = ((unsigned)((ga >> 32) & 0x01FFFFFFu)) | 0x80000000u;
    __builtin_amdgcn_tensor_load_to_lds(g0, g1, gz4, gz4, gz8, 0);
    const unsigned long long gb = (unsigned long long)(uintptr_t)bhh;
    u32x4 g0b;
    g0b.x = 1u;
    g0b.y = 2048u;                               // second half of bias_lds
    g0b.z = (unsigned)(gb & 0xFFFFFFFFu);
    g0b.w = ((unsigned)((gb >> 32) & 0x01FFFFFFu)) | 0x80000000u;
    __builtin_amdgcn_tensor_load_to_lds(g0b, g1, gz4, gz4, gz8, 0);
    __builtin_amdgcn_s_wait_tensorcnt(0);
  }
#else
  for (int i = threadIdx.x; i < 512; i += blockDim.x) {
    bias_lds[i] = bih[i];
    bias_lds[512 + i] = bhh[i];
  }
#endif

  // cooperative A-tile stage
  for (int idx = threadIdx.x; idx < 16 * (HIDC / 4); idx += blockDim.x) {
    const int rr = idx >> 5;
    const int cc = idx & 31;
    float4 v = *((const float4*)(inp + (size_t)(row0 + rr) * HIDC) + cc);
    *(float4*)&As[rr * A_STRIDE + cc * 4] = v;
  }
  __syncthreads();

  const int c0 = (wave << 4) + r;              // gate cols 0..255
  const int c1 = c0 + 256;                     // gate cols 256..511
  const float* arow  = &As[r * A_STRIDE];
  const float* b0row = wih + (size_t)c0 * HIDC;  // B[k][c] = wih[c][k]
  const float* b1row = wih + (size_t)c1 * HIDC;
  v8f acc0 = {};
  v8f acc1 = {};
#pragma unroll
  for (int kb = 0; kb < HIDC; kb += 4) {
    const int k0 = kb + (h2 << 1);
    v2f a = *(const v2f*)&arow[k0];
    v2f b0; b0.x = b0row[k0]; b0.y = b0row[k0 + 1];
    v2f b1; b1.x = b1row[k0]; b1.y = b1row[k0 + 1];
    acc0 = __builtin_amdgcn_wmma_f32_16x16x4_f32(false, a, false, b0, (short)0,
                                                 acc0, false, false);
    acc1 = __builtin_amdgcn_wmma_f32_16x16x4_f32(false, a, false, b1, (short)0,
                                                 acc1, false, false);
  }
  const float bias0 = bias_lds[c0] + bias_lds[512 + c0];
  const float bias1 = bias_lds[c1] + bias_lds[512 + c1];
#pragma unroll
  for (int i = 0; i < 8; ++i) {
    const int row = i + (h2 << 3);
    gl[row * LDS_STRIDE + c0] = acc0[i] + bias0;
    gl[row * LDS_STRIDE + c1] = acc1[i] + bias1;
  }
  __syncthreads();
  // 16 rows x 128 outputs = 2048 elements, 512 threads -> 4 each
  for (int t = threadIdx.x; t < 16 * HIDC; t += blockDim.x) {
    const int m = t >> 7;
    const int j = t & (HIDC - 1);
    const float iv = gl[m * LDS_STRIDE + j];
    const float gv = gl[m * LDS_STRIDE + j + 256];
    const float ov = gl[m * LDS_STRIDE + j + 384];
    const float c  = sigmoidf_(iv) * tanhf(gv);
    out[(size_t)(row0 + m) * HIDC + j] = sigmoidf_(ov) * tanhf(c);
  }
}

// ---------------------------------------------------------------------------
// fp32 WMMA GEMM: C[rows x OC] = A[rows x 128] @ B[128 x OC] (+ bias)
// OC is a template parameter so every B load / C store becomes
// base + compile-time offset (no v_mul_u64 in the hot loop).
// A tile (16x128) staged once in LDS; one wave per 16x16 output tile.
// blockDim.x = 32 * (OC/16); gridDim.x = rows/16.
// ---------------------------------------------------------------------------
template <int OC>
__global__ void gemm_wmma_f32(const float* __restrict__ A,
                              const float* __restrict__ B,
                              const float* __restrict__ bias,
                              float* __restrict__ C) {
  __shared__ float As[16 * A_STRIDE];
  const int lane = threadIdx.x & 31;
  const int wave = threadIdx.x >> 5;
  const int row0 = blockIdx.x << 4;
  const int col0 = wave << 4;
  const int r    = lane & 15;   // A row-in-tile / B,C col-in-tile
  const int h2   = lane >> 4;   // half-wave selector

  // cooperative A-tile stage: 16 rows x 32 float4
  for (int idx = threadIdx.x; idx < 16 * (HIDC / 4); idx += blockDim.x) {
    const int rr = idx >> 5;
    const int cc = idx & 31;
    float4 v = *((const float4*)(A + (size_t)(row0 + rr) * HIDC) + cc);
    *(float4*)&As[rr * A_STRIDE + cc * 4] = v;   // ds_store_b128 (16B aligned)
  }
  __syncthreads();

  v8f acc = {};
  const float* arow = &As[r * A_STRIDE];
  const float* bcol = B + col0 + r;              // per-lane B column base
#pragma unroll
  for (int kb = 0; kb < HIDC; kb += 4) {
    const int k0 = kb + (h2 << 1);
    v2f a = *(const v2f*)&arow[k0];              // ds_load_b64, conflict-free
    v2f b;                                       // B 4x16 row-striped across lanes
    b.x = bcol[k0 * OC];                         // constant-offset global loads
    b.y = bcol[(k0 + 1) * OC];
    acc = __builtin_amdgcn_wmma_f32_16x16x4_f32(false, a, false, b, (short)0,
                                                acc, false, false);
  }
  const float bv = bias ? bias[col0 + r] : 0.0f;
  float* crow = C + (size_t)row0 * OC + col0 + r;
#pragma unroll
  for (int i = 0; i < 8; ++i) {                  // C vgpr i: M=i / M=i+8
    crow[(i + (h2 << 3)) * OC] = acc[i] + bv;
  }
}

// ---------------------------------------------------------------------------
// Degree / normalization
// ---------------------------------------------------------------------------
__global__ void zero_f32_kernel(float* __restrict__ p, int n) {
  int i = blockIdx.x * blockDim.x + threadIdx.x;
  if (i < n) p[i] = 0.0f;
}

__global__ void deg_count_kernel(const int* __restrict__ dst,
                                 float* __restrict__ deg, int e) {
  int i = blockIdx.x * blockDim.x + threadIdx.x;
  if (i < e) atomicAdd(&deg[dst[i]], 1.0f);
}

__global__ void dinv_kernel(float* __restrict__ deg, int n) {
  int i = blockIdx.x * blockDim.x + threadIdx.x;
  if (i < n) deg[i] = rsqrtf(deg[i] + 1.0f);   // deg buffer becomes dinv
}

// ---------------------------------------------------------------------------
// GCN self-loop init: acc[n][c] = lin[n][c] * dinv[n]^2
// ---------------------------------------------------------------------------
__global__ void gcn_self_kernel(const float* __restrict__ lin,
                                const float* __restrict__ dinv,
                                float* __restrict__ acc, int total) {
  int i = blockIdx.x * blockDim.x + threadIdx.x;
  if (i < total) {
    int n = i >> 7;
    float d = dinv[n];
    acc[i] = lin[i] * d * d;
  }
}

// ---------------------------------------------------------------------------
// Edge scatter: one edge per wave32, 4 channels per lane. Grid-stride loop
// with global_prefetch of the NEXT iteration's gathered row to hide the
// random-access latency that dominates this (memory-roofline) phase.
// ---------------------------------------------------------------------------
__global__ void gcn_scatter_kernel(const float* __restrict__ lin,
                                   const float* __restrict__ dinv,
                                   const int* __restrict__ src,
                                   const int* __restrict__ dst,
                                   float* __restrict__ acc, int e) {
  const int total  = e << 5;                     // 32 lanes per edge
  const int stride = gridDim.x * blockDim.x;
  for (int t = blockIdx.x * blockDim.x + threadIdx.x; t < total; t += stride) {
    const int tn = t + stride;
    if (tn < total) {                            // prefetch next row (gfx1250 global_prefetch_b8)
      const int sn = src[tn >> 5];
      __builtin_prefetch(lin + (size_t)sn * HIDC + (tn & 31) * 4, 0, 1);
    }
    const int edge = t >> 5;
    const int lane = t & 31;
    const int s = src[edge];
    const int d = dst[edge];
    const float norm = dinv[s] * dinv[d];
    float4 v = *((const float4*)(lin + (size_t)s * HIDC) + lane);
    float* out = acc + (size_t)d * HIDC + lane * 4;
    atomicAdd(out + 0, v.x * norm);
    atomicAdd(out + 1, v.y * norm);
    atomicAdd(out + 2, v.z * norm);
    atomicAdd(out + 3, v.w * norm);
  }
}

// ---------------------------------------------------------------------------
// GCN finalize (in place): acc = relu(acc + bias[c])
// ---------------------------------------------------------------------------
__global__ void gcn_final_kernel(float* __restrict__ acc,
                                 const float* __restrict__ bias, int total) {
  int i = blockIdx.x * blockDim.x + threadIdx.x;
  if (i < total) {
    float v = acc[i] + bias[i & (HIDC - 1)];
    acc[i] = v > 0.0f ? v : 0.0f;
  }
}

// ---------------------------------------------------------------------------
// Host-side orchestration
// ---------------------------------------------------------------------------
static inline size_t align256(size_t x) { return (x + 255) & ~(size_t)255; }

extern "C" void kernel_launch(void* const* d_in, const int* in_sizes, int n_in,
                              void* d_out, int out_size, void* d_ws, size_t ws_size,
                              hipStream_t stream) {
  (void)n_in; (void)out_size; (void)ws_size;
  const int N = in_sizes[0] / HIDC;      // 50000
  const int E = in_sizes[1] / 2;         // 800000

  const float* x     = (const float*)d_in[0];
  const int*   edge  = (const int*)d_in[1];
  const int*   src   = edge;             // edge_index[0]
  const int*   dst   = edge + E;         // edge_index[1]
  const float* gw[3] = {(const float*)d_in[2], (const float*)d_in[4], (const float*)d_in[6]};
  const float* gb[3] = {(const float*)d_in[3], (const float*)d_in[5], (const float*)d_in[7]};
  const float* wih   = (const float*)d_in[8];    // [3][512][128]
  // d_in[9] = whh: unused (h0 == 0)
  const float* bih   = (const float*)d_in[10];   // [3][512]
  const float* bhh   = (const float*)d_in[11];   // [3][512]
  const float* lin_w = (const float*)d_in[12];   // [128][64]
  const float* lin_b = (const float*)d_in[13];   // [64]
  float*       outp  = (float*)d_out;            // [N][64]

  // workspace: dinv | lin buffer | acc buffer
  char* ws = (char*)d_ws;
  size_t off = 0;
  float* dinv = (float*)(ws + off); off += align256((size_t)N * sizeof(float));
  float* lin  = (float*)(ws + off); off += align256((size_t)N * HIDC * sizeof(float));
  float* acc  = (float*)(ws + off);

  const int total   = N * HIDC;
  const int rowTile = N / 16;            // 3125 (exact)

  // 1) degree -> dinv
  zero_f32_kernel<<<(N + 255) / 256, 256, 0, stream>>>(dinv, N);
  deg_count_kernel<<<(E + 255) / 256, 256, 0, stream>>>(dst, dinv, E);
  dinv_kernel<<<(N + 255) / 256, 256, 0, stream>>>(dinv, N);

  // 2) three GCN layers
  for (int l = 0; l < 3; ++l) {
    const float* h_in = (l == 0) ? x : acc;
    gemm_wmma_f32<HIDC><<<rowTile, 256, 0, stream>>>(h_in, gw[l], nullptr, lin);
    gcn_self_kernel<<<(total + 255) / 256, 256, 0, stream>>>(lin, dinv, acc, total);
    gcn_scatter_kernel<<<4096, 256, 0, stream>>>(lin, dinv, src, dst, acc, E);
    gcn_final_kernel<<<(total + 255) / 256, 256, 0, stream>>>(acc, gb[l], total);
  }

  // 3) three fused LSTM layers (ping-pong acc <-> lin)
  float* cur = acc;
  float* oth = lin;
  for (int l = 0; l < 3; ++l) {
    lstm_layer_kernel<<<rowTile, 512, 0, stream>>>(
        cur, wih + (size_t)l * 512 * HIDC, bih + (size_t)l * 512,
        bhh + (size_t)l * 512, oth);
    float* t = cur; cur = oth; oth = t;
  }

  // 4) final linear into d_out
  gemm_wmma_f32<OUTC><<<rowTile, 128, 0, stream>>>(cur, lin_w, lin_b, outp);
}